// GroupEmbedding_72980084294360
// MI455X (gfx1250) — compile-verified
//
#include <hip/hip_runtime.h>
#include <hip/hip_bf16.h>

typedef __attribute__((ext_vector_type(2))) float v2f;
typedef __attribute__((ext_vector_type(8))) float v8f;

#define GE_U 16
#define GE_B 50
#define GE_D 64
#define GE_SS 66   // padded LDS row stride (floats) to avoid bank conflicts
#define GE_FACTOR 0.5f

// One 16x64 = A[16x64] @ W^T + bias GEMM done by a single wave with
// v_wmma_f32_16x16x4_f32. A is in LDS (row stride GE_SS), W is [64][64]
// row-major in global (W^T applied via indexing), bias is [64].
// Result: acc[t] holds the 16x16 tile with N in [16t, 16t+16).
// C/D layout: VGPR r <-> M = r + 8*(lane>=16), N = 16t + (lane&15).
__device__ __forceinline__ void ge_gemm16x64(const float* __restrict__ A_lds,
                                             const float* __restrict__ W,
                                             const float* __restrict__ bias,
                                             v8f acc[4]) {
    const int lane = threadIdx.x & 31;
    const int l16  = lane & 15;
    const int koff = (lane >> 4) * 2;   // A/B frag: lanes 0-15 -> K=0,1 ; lanes 16-31 -> K=2,3
#pragma unroll
    for (int t = 0; t < 4; ++t) {
        const int n0 = t * 16;
        const float bsc = bias[n0 + l16];
        v8f c;
#pragma unroll
        for (int r = 0; r < 8; ++r) c[r] = bsc;   // C[u][n] starts at bias[n]
        const float* wrow = W + (size_t)(n0 + l16) * GE_D;
#pragma unroll
        for (int k0 = 0; k0 < GE_D; k0 += 4) {
            v2f a, b;
            a.x = A_lds[(size_t)l16 * GE_SS + k0 + koff];
            a.y = A_lds[(size_t)l16 * GE_SS + k0 + koff + 1];
            b.x = wrow[k0 + koff];          // B[k][n] = W[n][k]
            b.y = wrow[k0 + koff + 1];
            c = __builtin_amdgcn_wmma_f32_16x16x4_f32(
                    false, a, false, b, (short)0, c, false, false);
        }
        acc[t] = c;
    }
}

__global__ __launch_bounds__(256)
void GroupEmbedding_72980084294360_kernel(
        const int* __restrict__ group_users,   // [G,16]
        const int* __restrict__ user_lens,     // [G]
        const int* __restrict__ beh_ids,       // [G,16,50]
        const float* __restrict__ beh_counts,  // [G,16,50]
        const int* __restrict__ target_user,   // [G]
        const float* __restrict__ user_table,  // [N,64]
        const float* __restrict__ item_table,  // [N,64]
        const float* __restrict__ sim_vec,     // [N,64]
        const float* __restrict__ Wq, const float* __restrict__ bq,
        const float* __restrict__ Wk, const float* __restrict__ bk,
        const float* __restrict__ Wv, const float* __restrict__ bv,
        float* __restrict__ out)               // [G,64]
{
    __shared__ float sim_s [GE_U * GE_SS];   // similarity rows of the 16 members
    __shared__ float pers_s[GE_U * GE_SS];   // u_emb, then += behavior gather-sum
    __shared__ float K_s   [GE_U * GE_SS];   // K projection
    __shared__ float simT_s[GE_D];           // target similarity row
    __shared__ float Q_s   [GE_D];           // Q projection
    __shared__ float w_s   [GE_U];           // softmax weights

    const int g    = blockIdx.x;
    const int tid  = threadIdx.x;
    const int lane = tid & 31;
    const int wave = tid >> 5;

    // ---- Phase 1: stage member sim rows + user embeddings (+ target sim) ----
    {
        const int u  = tid >> 4;              // 0..15
        const int c4 = (tid & 15) * 4;        // 0,4,...,60
        const int uid = group_users[g * GE_U + u];
        const float4 s4 = *(const float4*)(sim_vec    + (size_t)uid * GE_D + c4);
        const float4 e4 = *(const float4*)(user_table + (size_t)uid * GE_D + c4);
        float* sp = sim_s  + u * GE_SS + c4;
        float* pp = pers_s + u * GE_SS + c4;
        sp[0] = s4.x; sp[1] = s4.y; sp[2] = s4.z; sp[3] = s4.w;
        pp[0] = e4.x; pp[1] = e4.y; pp[2] = e4.z; pp[3] = e4.w;
        if (tid < GE_D) {
            const int tgt = target_user[g];
            simT_s[tid] = sim_vec[(size_t)tgt * GE_D + tid];
        }
    }
    __syncthreads();

    // ---- Phase 2: ragged behavior gather-sum (dominant memory traffic) ----
    // wave w owns users 2w, 2w+1; each lane owns dims {2*lane, 2*lane+1}.
    {
#pragma unroll
        for (int uu = 0; uu < 2; ++uu) {
            const int u = wave * 2 + uu;
            const size_t base = ((size_t)g * GE_U + u) * GE_B;
            float ax = 0.f, ay = 0.f;
#pragma unroll 2
            for (int b = 0; b < GE_B; ++b) {
                const float cnt = beh_counts[base + b];
                if (cnt != 0.f) {                      // wave-uniform skip of padded slots
                    const int id = beh_ids[base + b];
                    const float2 r2 = *(const float2*)(item_table +
                                        (size_t)id * GE_D + lane * 2);
                    ax = fmaf(cnt, r2.x, ax);
                    ay = fmaf(cnt, r2.y, ay);
                }
            }
            pers_s[u * GE_SS + lane * 2 + 0] += ax;    // personalized = beh_sum + u_emb
            pers_s[u * GE_SS + lane * 2 + 1] += ay;
        }
    }
    __syncthreads();

    // ---- Phase 3: wave-specialized projections ----
    v8f accV[4];
    if (wave == 0) {                          // K = sim @ Wk^T + bk  -> LDS
        v8f accK[4];
        ge_gemm16x64(sim_s, Wk, bk, accK);
        const int l16 = lane & 15;
        const int mh  = (lane >> 4) * 8;
#pragma unroll
        for (int t = 0; t < 4; ++t)
#pragma unroll
            for (int r = 0; r < 8; ++r)
                K_s[(size_t)(r + mh) * GE_SS + t * 16 + l16] = accK[t][r];
    } else if (wave == 1) {                   // V = pers @ Wv^T + bv -> registers
        ge_gemm16x64(pers_s, Wv, bv, accV);
    } else if (wave < 4) {                    // Q[d] = bq[d] + sum_k simT[k]*Wq[d][k]
        const int d = tid - 64;
        float q = bq[d];
        const float* wr = Wq + (size_t)d * GE_D;
#pragma unroll 8
        for (int k = 0; k < GE_D; ++k) q = fmaf(simT_s[k], wr[k], q);
        Q_s[d] = q;
    }
    __syncthreads();

    // ---- Phase 4: masked softmax over the 16 members (lanes 0-15) ----
    if (tid < GE_U) {
        const int ulen = user_lens[g];
        float lg = 0.f;
#pragma unroll 8
        for (int d = 0; d < GE_D; ++d)
            lg = fmaf(Q_s[d], K_s[(size_t)tid * GE_SS + d], lg);
        lg *= GE_FACTOR;
        if (tid >= ulen) lg = -INFINITY;
        float m = lg;
#pragma unroll
        for (int off = 8; off > 0; off >>= 1) m = fmaxf(m, __shfl_xor(m, off, 16));
        const float e = __expf(lg - m);
        float s = e;
#pragma unroll
        for (int off = 8; off > 0; off >>= 1) s += __shfl_xor(s, off, 16);
        w_s[tid] = e / s;
    }
    __syncthreads();

    // ---- Phase 5: attention-weighted pooling of V (wave 1 holds V in regs) ----
    if (wave == 1) {
        const int mh = (lane >> 4) * 8;
#pragma unroll
        for (int t = 0; t < 4; ++t) {
            float s = 0.f;
#pragma unroll
            for (int r = 0; r < 8; ++r) s = fmaf(w_s[r + mh], accV[t][r], s);
            s += __shfl_down(s, 16, 32);      // fold M=8..15 half onto M=0..7 half
            if (lane < 16) out[(size_t)g * GE_D + t * 16 + lane] = s;
        }
    }
}

extern "C" void kernel_launch(void* const* d_in, const int* in_sizes, int n_in,
                              void* d_out, int out_size, void* d_ws, size_t ws_size,
                              hipStream_t stream) {
    const int*   group_users = (const int*)  d_in[0];
    const int*   user_lens   = (const int*)  d_in[1];
    const int*   beh_ids     = (const int*)  d_in[2];
    const float* beh_counts  = (const float*)d_in[3];
    const int*   target_user = (const int*)  d_in[4];
    const float* user_table  = (const float*)d_in[5];
    const float* item_table  = (const float*)d_in[6];
    const float* sim_vec     = (const float*)d_in[7];
    const float* Wq = (const float*)d_in[8];
    const float* bq = (const float*)d_in[9];
    const float* Wk = (const float*)d_in[10];
    const float* bk = (const float*)d_in[11];
    const float* Wv = (const float*)d_in[12];
    const float* bv = (const float*)d_in[13];
    float* out = (float*)d_out;

    const int G = in_sizes[1];   // user_lens has G elements
    GroupEmbedding_72980084294360_kernel<<<G, 256, 0, stream>>>(
        group_users, user_lens, beh_ids, beh_counts, target_user,
        user_table, item_table, sim_vec, Wq, bq, Wk, bk, Wv, bv, out);
}